// Mamba2Mixer_81243601371869
// MI455X (gfx1250) — compile-verified
//
#include <hip/hip_runtime.h>
#include <hip/hip_bf16.h>

typedef __attribute__((ext_vector_type(2))) float v2f;
typedef __attribute__((ext_vector_type(8))) float v8f;

#define D_MODEL   1024
#define D_STATE   128
#define D_CONV    4
#define HEADDIM   64
#define D_INNER   2048
#define NHEADS    32
#define CONV_DIM  2304
#define D_IN_PROJ 4384
#define BATCH     2
#define SEQLEN    1024
#define EPS       1e-5f

// GEMM tiling
#define BM    128     // block tile M
#define BN    128     // block tile N
#define KC    32      // K chunk staged in LDS
#define LDSTR 36      // padded LDS row stride (dwords): rows 16B-aligned, no bank conflicts

// ---------------------------------------------------------------------------
// C[M,N] = A[M,K] * W[N,K]^T (fp32, row-major) via v_wmma_f32_16x16x4_f32.
// 256 threads = 8 waves as 4(M) x 2(N); wave tile 32M x 64N (8 accumulators).
// Double-buffered LDS staging: global b128 loads for chunk k+1 issue before the
// 64 WMMAs of chunk k, so s_wait_loadcnt lands after the matrix work.
// All branches around WMMA are wave-uniform (EXEC must be all ones).
// ---------------------------------------------------------------------------
__global__ __launch_bounds__(256)
void gemm_f32_wmma(const float* __restrict__ A, const float* __restrict__ W,
                   float* __restrict__ C, int M, int N, int K) {
    __shared__ float ldsA[2][BM * LDSTR];
    __shared__ float ldsB[2][BN * LDSTR];

    const int tid   = threadIdx.x;
    const int lane  = tid & 31;
    const int wave  = tid >> 5;
    const int waveM = wave >> 1;          // 0..3 -> 32-row strip
    const int waveN = wave & 1;           // 0..1 -> 64-col strip
    const int row16 = lane & 15;
    const int kHalf = (lane >> 4) * 2;    // 0 or 2

    const int m0 = blockIdx.y * BM;
    const int n0 = blockIdx.x * BN;

    float4 pa[4], pb[4];                  // staged registers (global -> LDS)

    const v8f vzero = {0.f, 0.f, 0.f, 0.f, 0.f, 0.f, 0.f, 0.f};
    v8f acc[2][4];
#pragma unroll
    for (int mi = 0; mi < 2; ++mi)
#pragma unroll
        for (int j = 0; j < 4; ++j) acc[mi][j] = vzero;

    // ---- staging helpers: e = pp*256 + tid ; row = e/8, seg = e%8 (16B each)
    auto load_chunk = [&](int kc) {
#pragma unroll
        for (int pp = 0; pp < 4; ++pp) {
            int e = pp * 256 + tid, r = e >> 3, s = e & 7;
            pa[pp] = *(const float4*)(A + (size_t)(m0 + r) * K + kc + s * 4);
            int nb = n0 + r; if (nb > N - 1) nb = N - 1;   // clamp: keep loads in-range
            pb[pp] = *(const float4*)(W + (size_t)nb * K + kc + s * 4);
        }
    };
    auto store_chunk = [&](int buf) {
#pragma unroll
        for (int pp = 0; pp < 4; ++pp) {
            int e = pp * 256 + tid, r = e >> 3, s = e & 7;
            *(float4*)&ldsA[buf][r * LDSTR + s * 4] = pa[pp];
            *(float4*)&ldsB[buf][r * LDSTR + s * 4] = pb[pp];
        }
    };
    auto compute = [&](int buf) {
#pragma unroll
        for (int k4 = 0; k4 < KC; k4 += 4) {
            v2f a0 = *(const v2f*)&ldsA[buf][(waveM * 32 +      row16) * LDSTR + k4 + kHalf];
            v2f a1 = *(const v2f*)&ldsA[buf][(waveM * 32 + 16 + row16) * LDSTR + k4 + kHalf];
#pragma unroll
            for (int j = 0; j < 4; ++j) {
                v2f b = *(const v2f*)&ldsB[buf][(waveN * 64 + j * 16 + row16) * LDSTR + k4 + kHalf];
                acc[0][j] = __builtin_amdgcn_wmma_f32_16x16x4_f32(
                    false, a0, false, b, (short)0, acc[0][j], false, false);
                acc[1][j] = __builtin_amdgcn_wmma_f32_16x16x4_f32(
                    false, a1, false, b, (short)0, acc[1][j], false, false);
            }
        }
    };

    // ---- software pipeline over K chunks
    load_chunk(0);
    store_chunk(0);
    __syncthreads();
    int buf = 0;
    for (int kc = KC;; kc += KC) {
        const bool more = (kc < K);
        if (more) load_chunk(kc);       // global loads in flight during compute
        compute(buf);
        if (!more) break;
        store_chunk(buf ^ 1);           // waits loadcnt here, after the WMMAs
        __syncthreads();
        buf ^= 1;
    }

    // ---- store: vgpr r -> m = r + 8*(lane/16); n = lane%16 within each tile
    const int mBase = m0 + waveM * 32 + (lane >> 4) * 8;
#pragma unroll
    for (int mi = 0; mi < 2; ++mi) {
#pragma unroll
        for (int j = 0; j < 4; ++j) {
            int ntile = n0 + waveN * 64 + j * 16;
            if (ntile >= N) continue;   // wave-uniform (N is a multiple of 16)
            float* Cj = C + (size_t)(mBase + mi * 16) * N + ntile + row16;
#pragma unroll
            for (int r = 0; r < 8; ++r) Cj[(size_t)r * N] = acc[mi][j][r];
        }
    }
}

// ---------------------------------------------------------------------------
// Depthwise causal conv (width 4) + bias + SiLU over the xBC slice of zxbcdt.
// ---------------------------------------------------------------------------
__global__ __launch_bounds__(256)
void conv_silu_kernel(const float* __restrict__ zxbcdt,
                      const float* __restrict__ conv_w,
                      const float* __restrict__ conv_b,
                      float* __restrict__ xbc_out) {
    int idx = blockIdx.x * blockDim.x + threadIdx.x;
    const int total = BATCH * SEQLEN * CONV_DIM;
    if (idx >= total) return;
    int c = idx % CONV_DIM;
    int l = (idx / CONV_DIM) % SEQLEN;
    int b = idx / (CONV_DIM * SEQLEN);

    const float* src = zxbcdt + (size_t)b * SEQLEN * D_IN_PROJ + D_INNER + c;
    float acc = conv_b[c];
#pragma unroll
    for (int j = 0; j < D_CONV; ++j) {
        int ls = l - (D_CONV - 1) + j;
        if (ls >= 0) acc += src[(size_t)ls * D_IN_PROJ] * conv_w[c * D_CONV + j];
    }
    acc = acc / (1.0f + expf(-acc));             // SiLU
    xbc_out[idx] = acc;
}

// ---------------------------------------------------------------------------
// Sequential SSM scan. One block per (batch, head); 256 threads.
// State h is 64(p) x 128(n); thread (p = tid/4, q = tid%4) owns h[p][q*32..+31]
// in registers. B/C/x staged through LDS each step; y reduced across the quad.
// ---------------------------------------------------------------------------
__global__ __launch_bounds__(256)
void ssm_scan_kernel(const float* __restrict__ zxbcdt,
                     const float* __restrict__ xbc,
                     const float* __restrict__ dt_bias,
                     const float* __restrict__ A_log,
                     const float* __restrict__ Dvec,
                     float* __restrict__ y) {
    const int bh  = blockIdx.x;
    const int b   = bh >> 5;      // / NHEADS
    const int hd  = bh & 31;      // % NHEADS
    const int tid = threadIdx.x;
    const int p   = tid >> 2;
    const int q   = tid & 3;

    __shared__ float sB[D_STATE];
    __shared__ float sC[D_STATE];
    __shared__ float sx[HEADDIM];
    __shared__ float sdA, sdtp;

    const float Ah   = -expf(A_log[hd]);
    const float bias = dt_bias[hd];
    const float Dh   = Dvec[hd];

    float h[32];
#pragma unroll
    for (int i = 0; i < 32; ++i) h[i] = 0.0f;

    const float* xbc_b = xbc + (size_t)b * SEQLEN * CONV_DIM;
    const float* zx_b  = zxbcdt + (size_t)b * SEQLEN * D_IN_PROJ;
    float* y_b = y + (size_t)b * SEQLEN * D_INNER + hd * HEADDIM;

    for (int l = 0; l < SEQLEN; ++l) {
        const float* row = xbc_b + (size_t)l * CONV_DIM;
        if (tid < D_STATE) sB[tid] = row[D_INNER + tid];
        else               sC[tid - D_STATE] = row[D_INNER + tid];
        if (tid < HEADDIM) sx[tid] = row[hd * HEADDIM + tid];
        if (tid == 0) {
            float v   = zx_b[(size_t)l * D_IN_PROJ + (D_INNER + CONV_DIM) + hd] + bias;
            float dtp = (v > 20.0f) ? v : log1pf(expf(v));   // softplus
            sdtp = dtp;
            sdA  = expf(dtp * Ah);
        }
        __syncthreads();

        const float dA    = sdA;
        const float xp    = sx[p];
        const float coeff = sdtp * xp;
        const float* Bq   = sB + q * 32;
        const float* Cq   = sC + q * 32;
        float yp = 0.0f;
#pragma unroll
        for (int i = 0; i < 32; ++i) {
            h[i] = dA * h[i] + coeff * Bq[i];
            yp  += Cq[i] * h[i];
        }
        yp += __shfl_down(yp, 2, 32);
        yp += __shfl_down(yp, 1, 32);
        if (q == 0) y_b[(size_t)l * D_INNER + p] = yp + Dh * xp;
        __syncthreads();
    }
}

// ---------------------------------------------------------------------------
// RMSNorm over D_INNER + silu(z) gate. One block per (b,l) row.
// ---------------------------------------------------------------------------
__global__ __launch_bounds__(256)
void rmsnorm_gate_kernel(const float* __restrict__ y,
                         const float* __restrict__ zxbcdt,
                         const float* __restrict__ norm_w,
                         float* __restrict__ yn) {
    const int rowi = blockIdx.x;                 // 0 .. BATCH*SEQLEN-1
    const int tid  = threadIdx.x;
    const float* yrow = y + (size_t)rowi * D_INNER;
    const float* zrow = zxbcdt + (size_t)rowi * D_IN_PROJ;  // z = first D_INNER

    float ss = 0.0f;
    for (int i = tid; i < D_INNER; i += 256) { float v = yrow[i]; ss += v * v; }

    __shared__ float red[256];
    red[tid] = ss;
    __syncthreads();
    for (int s = 128; s > 0; s >>= 1) {
        if (tid < s) red[tid] += red[tid + s];
        __syncthreads();
    }
    const float rs = rsqrtf(red[0] / (float)D_INNER + EPS);

    for (int i = tid; i < D_INNER; i += 256) {
        float z = zrow[i];
        float g = z / (1.0f + expf(-z));         // silu(z)
        yn[(size_t)rowi * D_INNER + i] = yrow[i] * rs * norm_w[i] * g;
    }
}

// ---------------------------------------------------------------------------
extern "C" void kernel_launch(void* const* d_in, const int* in_sizes, int n_in,
                              void* d_out, int out_size, void* d_ws, size_t ws_size,
                              hipStream_t stream) {
    const float* x       = (const float*)d_in[0];  // (B, L, D_MODEL)
    const float* Win     = (const float*)d_in[1];  // (D_IN_PROJ, D_MODEL)
    const float* conv_w  = (const float*)d_in[2];  // (CONV_DIM, D_CONV)
    const float* conv_b  = (const float*)d_in[3];  // (CONV_DIM,)
    const float* dt_bias = (const float*)d_in[4];  // (NHEADS,)
    const float* A_log   = (const float*)d_in[5];  // (NHEADS,)
    const float* Dvec    = (const float*)d_in[6];  // (NHEADS,)
    const float* norm_w  = (const float*)d_in[7];  // (D_INNER,)
    const float* Wout    = (const float*)d_in[8];  // (D_MODEL, D_INNER)
    float* out = (float*)d_out;                    // (B, L, D_MODEL)

    const size_t M = (size_t)BATCH * SEQLEN;       // 2048 rows
    float* ws      = (float*)d_ws;
    float* zxbcdt  = ws;                                   // M * 4384
    float* xbc     = zxbcdt + M * D_IN_PROJ;               // M * 2304
    float* ybuf    = xbc + M * CONV_DIM;                   // M * 2048
    float* ynbuf   = ybuf + M * D_INNER;                   // M * 2048

    // 1) in-proj GEMM: zxbcdt[M,4384] = x[M,1024] @ Win^T
    {
        int Mi = (int)M, Ni = D_IN_PROJ, Ki = D_MODEL;
        dim3 grid((Ni + BN - 1) / BN, Mi / BM);
        gemm_f32_wmma<<<grid, 256, 0, stream>>>(x, Win, zxbcdt, Mi, Ni, Ki);
    }
    // 2) depthwise conv + bias + SiLU
    {
        int total = BATCH * SEQLEN * CONV_DIM;
        conv_silu_kernel<<<(total + 255) / 256, 256, 0, stream>>>(zxbcdt, conv_w, conv_b, xbc);
    }
    // 3) sequential SSM scan
    {
        ssm_scan_kernel<<<BATCH * NHEADS, 256, 0, stream>>>(zxbcdt, xbc, dt_bias, A_log, Dvec, ybuf);
    }
    // 4) RMSNorm + silu(z) gate
    {
        rmsnorm_gate_kernel<<<(int)M, 256, 0, stream>>>(ybuf, zxbcdt, norm_w, ynbuf);
    }
    // 5) out-proj GEMM: out[M,1024] = yn[M,2048] @ Wout^T
    {
        int Mi = (int)M, Ni = D_MODEL, Ki = D_INNER;
        dim3 grid((Ni + BN - 1) / BN, Mi / BM);
        gemm_f32_wmma<<<grid, 256, 0, stream>>>(ynbuf, Wout, out, Mi, Ni, Ki);
    }
}